// MyModel_62208306316098
// MI455X (gfx1250) — compile-verified
//
#include <hip/hip_runtime.h>
#include <math.h>

typedef __attribute__((ext_vector_type(2))) float v2f;
typedef __attribute__((ext_vector_type(8))) float v8f;

#define NU_F 0.1f
#define BN_EPS_F 1e-5f

// ---------------------------------------------------------------------------
// SLayer for dim-0 diagrams: log-transform + gaussian binning, d=2, K=150.
// One block per batch row. Masked points are compacted away at staging time
// (mask is 0/1 and the reduction is order-independent), halving the exp work.
// Inner loop per lane: 1 ds_load_b64 (wave broadcast) + 5 VALU + 1 v_exp_f32.
// ---------------------------------------------------------------------------
__global__ __launch_bounds__(160)
void slayer0_kernel(const float* __restrict__ dgm,     // [B,P,2]
                    const int*   __restrict__ mask,    // [B,P]
                    const float* __restrict__ centers, // [K,2]
                    const float* __restrict__ sharp,   // [K,2]
                    float* __restrict__ out,           // [B,K]
                    int P, int Kc)
{
    __shared__ float2 pts[1024];
    __shared__ int    scnt;
    const int b = blockIdx.x;
    const float s = 0.70710678118654752f;

    if (threadIdx.x == 0) scnt = 0;
    __syncthreads();

    for (int p = threadIdx.x; p < P; p += blockDim.x) {
        if (mask[(size_t)b * P + p] != 0) {
            float b0 = dgm[((size_t)b * P + p) * 2 + 0];
            float d0 = dgm[((size_t)b * P + p) * 2 + 1];
            float x  = (b0 + d0) * s;
            float y  = (d0 - b0) * s;
            if (y <= NU_F) y = __logf(fmaxf(y, 1e-12f) * (1.0f / NU_F)) + NU_F;
            int idx = atomicAdd(&scnt, 1);
            pts[idx] = make_float2(x, y);
        }
    }
    __syncthreads();
    const int n = scnt;

    const int k = threadIdx.x;
    if (k < Kc) {
        float cx = centers[2 * k + 0];
        float cy = centers[2 * k + 1];
        float sx = sharp[2 * k + 0]; sx *= sx;
        float sy = sharp[2 * k + 1]; sy *= sy;
        float acc = 0.0f;
#pragma unroll 4
        for (int p = 0; p < n; ++p) {
            float2 pt = pts[p];
            float dx = pt.x - cx;
            float dy = pt.y - cy;
            float q  = fmaf(sx, dx * dx, sy * (dy * dy));
            acc += __expf(-q);
        }
        out[(size_t)b * Kc + k] = acc;
    }
}

// ---------------------------------------------------------------------------
// SLayer for essential diagrams: birth coordinate only, d=1, K=50, P=512.
// Same compaction trick.
// ---------------------------------------------------------------------------
__global__ __launch_bounds__(64)
void slayerE_kernel(const float* __restrict__ dgm,     // [B,P,2] (use [...,0])
                    const int*   __restrict__ mask,    // [B,P]
                    const float* __restrict__ centers, // [K,1]
                    const float* __restrict__ sharp,   // [K,1]
                    float* __restrict__ out,           // [B,K]
                    int P, int Kc)
{
    __shared__ float xs[512];
    __shared__ int   scnt;
    const int b = blockIdx.x;

    if (threadIdx.x == 0) scnt = 0;
    __syncthreads();

    for (int p = threadIdx.x; p < P; p += blockDim.x) {
        if (mask[(size_t)b * P + p] != 0) {
            int idx = atomicAdd(&scnt, 1);
            xs[idx] = dgm[((size_t)b * P + p) * 2 + 0];
        }
    }
    __syncthreads();
    const int n = scnt;

    const int k = threadIdx.x;
    if (k < Kc) {
        float c  = centers[k];
        float sk = sharp[k]; sk *= sk;
        float acc = 0.0f;
#pragma unroll 4
        for (int p = 0; p < n; ++p) {
            float dx = xs[p] - c;
            acc += __expf(-(sk * dx * dx));
        }
        out[(size_t)b * Kc + k] = acc;
    }
}

// ---------------------------------------------------------------------------
// Generic f32 WMMA GEMM: Y[:, ycol0 : ycol0+N] = epi(X[B,K] @ W[K,N] + bias)
// One wave per 16x16 output tile, contraction via V_WMMA_F32_16X16X4_F32.
// flags bit0 = BatchNorm (eval), bit1 = ReLU.
//
// Boundary handling is hoisted out of the hot loop:
//  - OOB columns: B operand junk only affects the un-stored D column -> just
//    clamp the column pointer once; guard only the stores.
//  - K remainder: zeroing B alone kills the k>=K terms for every row -> one
//    masked tail iteration; the main loop is pure load->wmma.
// ---------------------------------------------------------------------------
__global__ __launch_bounds__(32)
void gemm_epi_kernel(const float* __restrict__ X, int ldx,
                     const float* __restrict__ W,    // [K,N] row-major
                     const float* __restrict__ bias, // [N]
                     const float* __restrict__ g, const float* __restrict__ be,
                     const float* __restrict__ m, const float* __restrict__ v,
                     float* __restrict__ Y, int ldy, int ycol0,
                     int K, int N, int flags)
{
    const int lane = threadIdx.x;          // 0..31
    const int half = lane >> 4;            // 0 | 1
    const int l    = lane & 15;
    const int row0 = blockIdx.x * 16;
    const int n0   = blockIdx.y * 16;

    const int  col   = n0 + l;
    const bool colOK = (col < N);
    const int  colc  = colOK ? col : (N - 1);

    const float* __restrict__ xrow = X + (size_t)(row0 + l) * ldx;
    const float* __restrict__ wp   = W + colc;

    v8f acc = {0.f, 0.f, 0.f, 0.f, 0.f, 0.f, 0.f, 0.f};

    const int K4 = K & ~3;
    for (int k0 = 0; k0 < K4; k0 += 4) {
        const int ka = k0 + 2 * half;
        v2f a, bv;
        a[0]  = xrow[ka];
        a[1]  = xrow[ka + 1];
        bv[0] = wp[(size_t)ka * N];
        bv[1] = wp[(size_t)(ka + 1) * N];
        acc = __builtin_amdgcn_wmma_f32_16x16x4_f32(
                  false, a, false, bv, (short)0, acc, false, false);
    }
    if (K4 < K) {
        const int ka  = K4 + 2 * half;
        const int k0c = (ka     < K) ? ka     : (K - 1);
        const int k1c = (ka + 1 < K) ? ka + 1 : (K - 1);
        const float m0 = (ka     < K) ? 1.0f : 0.0f;
        const float m1 = (ka + 1 < K) ? 1.0f : 0.0f;
        v2f a, bv;
        a[0]  = xrow[k0c];
        a[1]  = xrow[k1c];
        bv[0] = wp[(size_t)k0c * N] * m0;
        bv[1] = wp[(size_t)k1c * N] * m1;
        acc = __builtin_amdgcn_wmma_f32_16x16x4_f32(
                  false, a, false, bv, (short)0, acc, false, false);
    }

    // epilogue (per-lane column is fixed -> BN params are lane-resident scalars)
    if (colOK) {
        float bb = bias[col], scale = 1.0f, shift = 0.0f;
        if (flags & 1) {
            float rs = rsqrtf(v[col] + BN_EPS_F);
            scale = g[col] * rs;
            shift = be[col] - m[col] * scale;
        }
#pragma unroll
        for (int r = 0; r < 8; ++r) {
            int   row = row0 + r + 8 * half;
            float val = acc[r] + bb;
            if (flags & 1) val = fmaf(val, scale, shift);
            if (flags & 2) val = fmaxf(val, 0.0f);
            Y[(size_t)row * ldy + ycol0 + col] = val;
        }
    }
}

// ---------------------------------------------------------------------------
extern "C" void kernel_launch(void* const* d_in, const int* in_sizes, int n_in,
                              void* d_out, int out_size, void* d_ws, size_t ws_size,
                              hipStream_t stream)
{
    (void)in_sizes; (void)n_in; (void)out_size; (void)ws_size;
    const int B = 1024;

    // --- input pointers: recursive insertion-order flattening of setup_inputs() ---
    const float* dgm0   = (const float*)d_in[0];
    const float* dgm0e  = (const float*)d_in[1];
    const float* dgm1e  = (const float*)d_in[2];
    const int*   mask0  = (const int*)d_in[3];
    const int*   mask0e = (const int*)d_in[4];
    const int*   mask1e = (const int*)d_in[5];
    const float* c0  = (const float*)d_in[6];   // centers0  [150,2]
    const float* s0  = (const float*)d_in[7];   // sharp0    [150,2]
    const float* c0e = (const float*)d_in[8];   // centers0e [50,1]
    const float* s0e = (const float*)d_in[9];
    const float* c1e = (const float*)d_in[10];
    const float* s1e = (const float*)d_in[11];
    // stage blocks s0,s1,s2: {w1,b1,w2,b2,g1,be1,m1,v1}
    const float* const* S0 = (const float* const*)&d_in[12];
    const float* const* S1 = (const float* const*)&d_in[20];
    const float* const* S2 = (const float* const*)&d_in[28];
    // head: {w1,b1,g1,be1,m1,v1, w2,b2,g2,be2,m2,v2, w3,..., w4,...}
    const float* const* H  = (const float* const*)&d_in[36];

    // --- workspace layout (floats) ---
    float* ws   = (float*)d_ws;
    float* x0   = ws;                 // [1024,150]
    float* x0e  = x0   + 1024 * 150;  // [1024,50]
    float* x1e  = x0e  + 1024 * 50;   // [1024,50]
    float* tmpA = x1e  + 1024 * 50;   // up to [1024,75] (reused per branch)
    float* xcat = tmpA + 1024 * 75;   // [1024,125]
    float* h1   = xcat + 1024 * 125;  // [1024,200]
    float* h2   = h1   + 1024 * 200;  // [1024,100]
    float* h3   = h2   + 1024 * 100;  // [1024,50]

    // --- SLayer stage (trans/VALU-bound; LDS-staged with mask compaction) ---
    slayer0_kernel<<<B, 160, 0, stream>>>(dgm0,  mask0,  c0,  s0,  x0,  1024, 150);
    slayerE_kernel<<<B,  64, 0, stream>>>(dgm0e, mask0e, c0e, s0e, x0e,  512,  50);
    slayerE_kernel<<<B,  64, 0, stream>>>(dgm1e, mask1e, c1e, s1e, x1e,  512,  50);

    // --- WMMA GEMM + fused epilogue launcher ---
    auto gemm = [&](const float* X, int ldx, const float* W, const float* bias,
                    const float* gg, const float* bb, const float* mm, const float* vv,
                    float* Y, int ldy, int ycol0, int K, int N, int flags) {
        dim3 grid(B / 16, (N + 15) / 16);
        gemm_epi_kernel<<<grid, 32, 0, stream>>>(X, ldx, W, bias, gg, bb, mm, vv,
                                                 Y, ldy, ycol0, K, N, flags);
    };

    // stage1 branch 0: 150 -> 75 (bias+BN) -> 75 (bias+ReLU) into xcat[:,0:75]
    gemm(x0,  150, S0[0], S0[1], S0[4], S0[5], S0[6], S0[7], tmpA,  75, 0, 150, 75, /*BN*/1);
    gemm(tmpA, 75, S0[2], S0[3], nullptr, nullptr, nullptr, nullptr, xcat, 125, 0, 75, 75, /*ReLU*/2);
    // stage1 branch 1: 50 -> 25 -> 25 into xcat[:,75:100]
    gemm(x0e,  50, S1[0], S1[1], S1[4], S1[5], S1[6], S1[7], tmpA,  25, 0,  50, 25, 1);
    gemm(tmpA, 25, S1[2], S1[3], nullptr, nullptr, nullptr, nullptr, xcat, 125, 75, 25, 25, 2);
    // stage1 branch 2: 50 -> 25 -> 25 into xcat[:,100:125]
    gemm(x1e,  50, S2[0], S2[1], S2[4], S2[5], S2[6], S2[7], tmpA,  25, 0,  50, 25, 1);
    gemm(tmpA, 25, S2[2], S2[3], nullptr, nullptr, nullptr, nullptr, xcat, 125, 100, 25, 25, 2);

    // head: 125 -> 200 -> 100 -> 50 -> 5
    gemm(xcat, 125, H[0],  H[1],  H[2],  H[3],  H[4],  H[5],  h1, 200, 0, 125, 200, 3);
    gemm(h1,   200, H[6],  H[7],  H[8],  H[9],  H[10], H[11], h2, 100, 0, 200, 100, 3);
    gemm(h2,   100, H[12], H[13], H[14], H[15], H[16], H[17], h3,  50, 0, 100,  50, 3);
    gemm(h3,    50, H[18], H[19], H[20], H[21], H[22], H[23], (float*)d_out, 5, 0, 50, 5, 1);
}